// ContextDisBlock_67800353734735
// MI455X (gfx1250) — compile-verified
//
#include <hip/hip_runtime.h>
#include <hip/hip_bf16.h>
#include <math.h>

// Problem constants
#define B_  8
#define C_  256
#define H_  56
#define W_  56
#define HW_ (H_ * W_)
#define PLANES_ 16          // C/RATIO
#define DENOM 6272.0f       // 2 * (max(H,W)/sigma)^2 = 2*56*56

// Padded GEMM size and LDS row stride (66 floats: avoids 64-bank column
// conflicts, keeps even-float offsets so v2f loads stay 8-byte aligned)
#define P_  64
#define LD_ 66

typedef __attribute__((ext_vector_type(2))) float v2f;
typedef __attribute__((ext_vector_type(8))) float v8f;

// ---------------------------------------------------------------------------
// 64x64x64 f32 GEMM on LDS tiles using V_WMMA_F32_16X16X4_F32 (wave32).
// D = A @ B, all matrices 64x64 stored in LDS with row stride LD_.
// Each of the 4 waves computes one 16-row stripe of D (4 tiles of 16x16).
//
// ISA layouts (CDNA5 ISA §7.12.2, 32-bit):
//   A 16x4 : lanes 0-15 -> M=lane, {K=k,k+1}; lanes 16-31 -> M=lane-16, {K=k+2,k+3}
//   B 4x16 : lanes 0-15 -> N=lane, rows {k,k+1}; lanes 16-31 -> N=lane-16, rows {k+2,k+3}
//   C/D    : VGPR r, lanes 0-15 -> M=r, N=lane; lanes 16-31 -> M=8+r, N=lane-16
// ---------------------------------------------------------------------------
__device__ __forceinline__ void wmma_gemm64(const float* __restrict__ A,
                                            const float* __restrict__ Bm,
                                            float* __restrict__ D,
                                            int wave, int lane) {
  const int m  = lane & 15;          // M (for A) / N (for B) within tile
  const int kh = (lane >> 4) << 1;   // 0 or 2: K sub-pair handled by this half-wave

  v8f acc0 = {}, acc1 = {}, acc2 = {}, acc3 = {};
  const int arow = (wave << 4) + m;

#pragma unroll
  for (int k = 0; k < P_; k += 4) {
    // A fragment: 2 consecutive K values from this lane's row
    v2f a = *(const v2f*)&A[arow * LD_ + k + kh];
    // B fragments for the 4 N-tiles (two rows of B, LD_ apart)
    const int br = (k + kh) * LD_;
    v2f b0 = { Bm[br + m],      Bm[br + LD_ + m]      };
    v2f b1 = { Bm[br + 16 + m], Bm[br + LD_ + 16 + m] };
    v2f b2 = { Bm[br + 32 + m], Bm[br + LD_ + 32 + m] };
    v2f b3 = { Bm[br + 48 + m], Bm[br + LD_ + 48 + m] };

    acc0 = __builtin_amdgcn_wmma_f32_16x16x4_f32(false, a, false, b0, (short)0, acc0, false, false);
    acc1 = __builtin_amdgcn_wmma_f32_16x16x4_f32(false, a, false, b1, (short)0, acc1, false, false);
    acc2 = __builtin_amdgcn_wmma_f32_16x16x4_f32(false, a, false, b2, (short)0, acc2, false, false);
    acc3 = __builtin_amdgcn_wmma_f32_16x16x4_f32(false, a, false, b3, (short)0, acc3, false, false);
  }

  // Scatter D per the C/D layout
  const int drow = (wave << 4) + ((lane >> 4) << 3);
#pragma unroll
  for (int r = 0; r < 8; ++r) {
    D[(drow + r) * LD_ + m]      = acc0[r];
    D[(drow + r) * LD_ + 16 + m] = acc1[r];
    D[(drow + r) * LD_ + 32 + m] = acc2[r];
    D[(drow + r) * LD_ + 48 + m] = acc3[r];
  }
}

// ---------------------------------------------------------------------------
// 1) cm[b,n] = sum_c x[b,c,n]*w_mask[c] + b_mask ;  m1[b,n] = |same w/ w_mask1|
//    Single pass over x, both dot products fused. Coalesced over n.
// ---------------------------------------------------------------------------
#define NBLK_N ((HW_ + 255) / 256)

__global__ __launch_bounds__(256) void mask_reduce_kernel(
    const float* __restrict__ x,
    const float* __restrict__ wm,  const float* __restrict__ bm,
    const float* __restrict__ wm1, const float* __restrict__ bm1,
    float* __restrict__ cm, float* __restrict__ m1) {
  __shared__ float swm[C_], swm1[C_];
  const int t = threadIdx.x;
  swm[t]  = wm[t];
  swm1[t] = wm1[t];
  __syncthreads();

  const int b = blockIdx.x / NBLK_N;
  const int n = (blockIdx.x % NBLK_N) * 256 + t;
  if (n < HW_) {
    const float* xp = x + (size_t)b * C_ * HW_ + n;
    float a0 = 0.f, a1 = 0.f;
#pragma unroll 4
    for (int c = 0; c < C_; ++c) {
      float v = xp[(size_t)c * HW_];
      a0 += v * swm[c];
      a1 += v * swm1[c];
    }
    cm[b * HW_ + n] = a0 + bm[0];
    m1[b * HW_ + n] = fabsf(a1 + bm1[0]);
  }
}

// ---------------------------------------------------------------------------
// 2) Per-batch: softmax(cm) in-place (stable), wexp = exp(m1 - max(m1)).
// ---------------------------------------------------------------------------
__global__ __launch_bounds__(256) void stats_kernel(
    float* __restrict__ cm, const float* __restrict__ m1,
    float* __restrict__ wexp) {
  __shared__ float red[256];
  const int b = blockIdx.x, t = threadIdx.x;
  float* cmb = cm + b * HW_;
  const float* m1b = m1 + b * HW_;

  float mx = -3.0e38f, mx1 = -3.0e38f;
  for (int n = t; n < HW_; n += 256) {
    mx  = fmaxf(mx,  cmb[n]);
    mx1 = fmaxf(mx1, m1b[n]);
  }
  red[t] = mx; __syncthreads();
  for (int s = 128; s > 0; s >>= 1) { if (t < s) red[t] = fmaxf(red[t], red[t + s]); __syncthreads(); }
  const float cmmax = red[0]; __syncthreads();

  red[t] = mx1; __syncthreads();
  for (int s = 128; s > 0; s >>= 1) { if (t < s) red[t] = fmaxf(red[t], red[t + s]); __syncthreads(); }
  const float m1max = red[0]; __syncthreads();

  float se = 0.f;
  for (int n = t; n < HW_; n += 256) se += __expf(cmb[n] - cmmax);
  red[t] = se; __syncthreads();
  for (int s = 128; s > 0; s >>= 1) { if (t < s) red[t] += red[t + s]; __syncthreads(); }
  const float inv = 1.f / red[0];

  for (int n = t; n < HW_; n += 256) {
    cmb[n] = __expf(cmb[n] - cmmax) * inv;           // softmax prob
    wexp[b * HW_ + n] = __expf(m1b[n] - m1max);      // spatial key weight
  }
}

// ---------------------------------------------------------------------------
// 3) ch_ctx[b,c] = sum_n x[b,c,n] * prob[b,n]   (one block per (b,c))
// ---------------------------------------------------------------------------
__global__ __launch_bounds__(256) void chctx_kernel(
    const float* __restrict__ x, const float* __restrict__ prob,
    float* __restrict__ chctx) {
  __shared__ float red[256];
  const int bc = blockIdx.x, t = threadIdx.x;
  const int b = bc >> 8;  // C_ == 256
  const float* xp = x + (size_t)bc * HW_;
  const float* pp = prob + b * HW_;
  float a = 0.f;
  for (int n = t; n < HW_; n += 256) a += xp[n] * pp[n];
  red[t] = a; __syncthreads();
  for (int s = 128; s > 0; s >>= 1) { if (t < s) red[t] += red[t + s]; __syncthreads(); }
  if (t == 0) chctx[bc] = red[0];
}

// ---------------------------------------------------------------------------
// 4) gate = sigmoid(relu(ch_ctx @ w_fc1^T) @ w_fc2^T)   (tiny MLP per batch)
// ---------------------------------------------------------------------------
__global__ __launch_bounds__(256) void gate_kernel(
    const float* __restrict__ chctx, const float* __restrict__ w_fc1,
    const float* __restrict__ w_fc2, float* __restrict__ gate) {
  __shared__ float ch[C_];
  __shared__ float hid[PLANES_];
  const int b = blockIdx.x, t = threadIdx.x;
  ch[t] = chctx[b * C_ + t];
  __syncthreads();
  if (t < PLANES_) {
    float h = 0.f;
    for (int c = 0; c < C_; ++c) h += w_fc1[t * C_ + c] * ch[c];
    hid[t] = fmaxf(h, 0.f);
  }
  __syncthreads();
  float g = 0.f;
#pragma unroll
  for (int k = 0; k < PLANES_; ++k) g += w_fc2[t * PLANES_ + k] * hid[k];
  gate[b * C_ + t] = 1.f / (1.f + __expf(-g));
}

// ---------------------------------------------------------------------------
// Build padded Gaussian matrix G[p][q] = exp(-(p-q)^2/6272), zero outside 56x56
// ---------------------------------------------------------------------------
__device__ __forceinline__ void build_G(float* sG, int tid, int nthreads) {
  for (int i = tid; i < P_ * P_; i += nthreads) {
    int p = i >> 6, q = i & 63;
    float g = 0.f;
    if (p < H_ && q < W_) {
      float d = (float)(p - q);
      g = __expf(-d * d * (1.0f / DENOM));
    }
    sG[p * LD_ + q] = g;
  }
}

// ---------------------------------------------------------------------------
// 5) Z[b] = G · W[b] · G   (softmax denominator plane; grid = B)
// ---------------------------------------------------------------------------
__global__ __launch_bounds__(128) void z_plane_kernel(
    const float* __restrict__ wexp, float* __restrict__ Zout) {
  __shared__ float sG[P_ * LD_];
  __shared__ float sA[P_ * LD_];
  __shared__ float sT[P_ * LD_];
  const int b = blockIdx.x;
  const int tid = threadIdx.x, wave = tid >> 5, lane = tid & 31;

  build_G(sG, tid, 128);
  const float* wp = wexp + (size_t)b * HW_;
  for (int i = tid; i < P_ * P_; i += 128) {
    int p = i >> 6, q = i & 63;
    sA[p * LD_ + q] = (p < H_ && q < W_) ? wp[p * W_ + q] : 0.f;
  }
  __syncthreads();
  wmma_gemm64(sA, sG, sT, wave, lane);   // T = W · G
  __syncthreads();
  wmma_gemm64(sG, sT, sA, wave, lane);   // Z = G · T   (overwrite sA)
  __syncthreads();
  for (int i = tid; i < HW_; i += 128)
    Zout[b * HW_ + i] = sA[(i / W_) * LD_ + (i % W_)];
}

// ---------------------------------------------------------------------------
// 6) Per-(b,c) plane: O = G · (x∘w) · G ; out = x + O/Z + x*gate   (grid = B*C)
// ---------------------------------------------------------------------------
__global__ __launch_bounds__(128) void ctx_plane_kernel(
    const float* __restrict__ x, const float* __restrict__ wexp,
    const float* __restrict__ Z, const float* __restrict__ gate,
    float* __restrict__ out) {
  __shared__ float sG[P_ * LD_];   // 16.5 KB
  __shared__ float sA[P_ * LD_];   // 16.5 KB (XW, then final O)
  __shared__ float sT[P_ * LD_];   // 16.5 KB
  __shared__ float sX[HW_];        // 12.25 KB raw x plane   (total ~61.8 KB)

  const int bc = blockIdx.x;
  const int b = bc >> 8;           // C_ == 256
  const int tid = threadIdx.x, wave = tid >> 5, lane = tid & 31;

  build_G(sG, tid, 128);
  const float* xp = x + (size_t)bc * HW_;
  const float* wp = wexp + (size_t)b * HW_;
  for (int i = tid; i < P_ * P_; i += 128) {
    int p = i >> 6, q = i & 63;
    float v = 0.f;
    if (p < H_ && q < W_) {
      float xv = xp[p * W_ + q];
      sX[p * W_ + q] = xv;
      v = xv * wp[p * W_ + q];
    }
    sA[p * LD_ + q] = v;
  }
  const float gt = gate[bc];
  __syncthreads();

  wmma_gemm64(sA, sG, sT, wave, lane);   // T = (x∘w) · G
  __syncthreads();
  wmma_gemm64(sG, sT, sA, wave, lane);   // O = G · T
  __syncthreads();

  const float* zp = Z + (size_t)b * HW_;
  float* op = out + (size_t)bc * HW_;
  for (int i = tid; i < HW_; i += 128) {
    float xv = sX[i];
    float ctx = sA[(i / W_) * LD_ + (i % W_)] / zp[i];
    op[i] = xv + ctx + xv * gt;          // x + context + channel_attention
  }
}

// ---------------------------------------------------------------------------
// Launcher
// ---------------------------------------------------------------------------
extern "C" void kernel_launch(void* const* d_in, const int* in_sizes, int n_in,
                              void* d_out, int out_size, void* d_ws, size_t ws_size,
                              hipStream_t stream) {
  const float* x      = (const float*)d_in[0];
  const float* w_mask = (const float*)d_in[1];
  const float* b_mask = (const float*)d_in[2];
  const float* w_mask1= (const float*)d_in[3];
  const float* b_mask1= (const float*)d_in[4];
  const float* w_fc1  = (const float*)d_in[5];
  const float* w_fc2  = (const float*)d_in[6];
  float* outp = (float*)d_out;

  float* ws    = (float*)d_ws;
  float* cm    = ws;                    // [B,HW] -> becomes softmax probs
  float* m1    = cm    + B_ * HW_;      // [B,HW]
  float* wexp  = m1    + B_ * HW_;      // [B,HW]
  float* Zbuf  = wexp  + B_ * HW_;      // [B,HW]
  float* chctx = Zbuf  + B_ * HW_;      // [B,C]
  float* gate  = chctx + B_ * C_;       // [B,C]

  mask_reduce_kernel<<<B_ * NBLK_N, 256, 0, stream>>>(x, w_mask, b_mask,
                                                      w_mask1, b_mask1, cm, m1);
  stats_kernel<<<B_, 256, 0, stream>>>(cm, m1, wexp);
  chctx_kernel<<<B_ * C_, 256, 0, stream>>>(x, cm, chctx);
  gate_kernel<<<B_, 256, 0, stream>>>(chctx, w_fc1, w_fc2, gate);
  z_plane_kernel<<<B_, 128, 0, stream>>>(wexp, Zbuf);
  ctx_plane_kernel<<<B_ * C_, 128, 0, stream>>>(x, wexp, Zbuf, gate, outp);
}